// HWTV_S_20409684590798
// MI455X (gfx1250) — compile-verified
//
#include <hip/hip_runtime.h>
#include <math.h>

#define NN 4
#define HH 128
#define WWD 128
#define CC 31
#define FIB (NN*HH*WWD)   // 65536 fibers along C
#define FPB 128           // fibers per block
#define STR 33            // padded LDS fiber stride (floats)
#define TPB 256           // 8 wave32 per block
#define ITE 20

typedef __attribute__((ext_vector_type(2))) float v2f;
typedef __attribute__((ext_vector_type(8))) float v8f;

__device__ __forceinline__ float bval(const float* __restrict__ mrow, int k, int n) {
    // B[k][n] = m[(n-k) mod 31] for k,n < 31, else 0 (padding row/col)
    int d = n - k;
    d += (d < 0) ? CC : 0;          // d in [0,31]; slot 31 of mrow is 0
    float v = mrow[d];
    return (k < CC && n < CC) ? v : 0.0f;
}

__launch_bounds__(TPB, 1)
__global__ void hwtv_admm_kernel(const float* __restrict__ gY,
                                 const float* __restrict__ gW,
                                 const float* __restrict__ gDemo,
                                 float* __restrict__ gOut) {
    __shared__ float sY [FPB*STR];
    __shared__ float sWt[FPB*STR];
    __shared__ float sX [FPB*STR];
    __shared__ float sZ [FPB*STR];   // holds rhs, then Z (safely aliased)
    __shared__ float sG1[FPB*STR];
    __shared__ float sG2[FPB*STR];
    __shared__ float sD [FPB*STR];   // DsZ
    __shared__ float sU [FPB*STR];   // Us
    __shared__ float sM [ITE*32];    // circulant-inverse kernel per iteration
    __shared__ float sDem[32];

    const int tid = threadIdx.x;
    const size_t base = (size_t)blockIdx.x * (size_t)(FPB*CC);

    if (tid < 32) sDem[tid] = (tid < CC) ? gDemo[tid] : 0.0f;

    // ---- load inputs, init state: X=Z=Y, G1=G2=0 ----
    for (int i = tid; i < FPB*CC; i += TPB) {
        int f = i / CC, c = i - f*CC;
        float y = gY[base + i];
        float w = gW[base + i];
        int o = f*STR + c;
        sY[o] = y; sWt[o] = w; sX[o] = y; sZ[o] = y;
        sG1[o] = 0.0f; sG2[o] = 0.0f;
    }
    for (int f = tid; f < FPB; f += TPB) { sZ[f*STR+31] = 0.0f; sZ[f*STR+32] = 0.0f; }
    __syncthreads();

    // DsZ = diff_h(Z) = roll(Z,-1) - Z (circular along C)
    for (int i = tid; i < FPB*CC; i += TPB) {
        int f = i / CC, c = i - f*CC;
        int cp = (c == CC-1) ? 0 : c+1;
        sD[f*STR+c] = sZ[f*STR+cp] - sZ[f*STR+c];
    }
    // ---- build circulant inverse kernels for all 20 iterations ----
    // m_it[c] = (1/31) * sum_k cos(2*pi*k*c/31) / (mu1 + mu2*demo[k])
    for (int e = tid; e < ITE*32; e += TPB) {
        int it = e >> 5, c = e & 31;
        if (c == 31) { sM[e] = 0.0f; continue; }
        double mud = 0.1 * pow(1.05, (double)it);  // mu1 == mu2 every iteration
        float fmu = (float)mud;
        float acc = 0.0f;
        for (int k = 0; k < CC; ++k) {
            int p = (k * c) % CC;
            float ang = 0.20268339700579309f * (float)p;   // 2*pi/31
            acc += cosf(ang) / (fmu + fmu * sDem[k]);
        }
        sM[e] = acc * (1.0f/31.0f);
    }
    __syncthreads();

    const int lane  = tid & 31;
    const int wv    = tid >> 5;
    const int laneM = lane & 15;
    const int hi    = lane >> 4;        // 0: lanes 0-15, 1: lanes 16-31

    double mud = 0.1;
    for (int it = 0; it < ITE; ++it) {
        const float mu1 = (float)mud;
        const float mu2 = (float)mud;

        // Phase A: Us = shrink(DsZ - G2/mu2, LAM/mu2)
        const float thr = 0.1f / mu2;
        for (int i = tid; i < FPB*CC; i += TPB) {
            int f = i / CC, c = i - f*CC; int o = f*STR+c;
            float v = sD[o] - sG2[o] / mu2;
            float a = fabsf(v) - thr; a = a > 0.0f ? a : 0.0f;
            sU[o] = (v >= 0.0f) ? a : -a;
        }
        __syncthreads();

        // Phase B: rhs = mu1*X + G1 + Dt(mu2*Us + G2); Dt(v)[c] = v[c-1]-v[c]
        for (int i = tid; i < FPB*CC; i += TPB) {
            int f = i / CC, c = i - f*CC;
            int cm = (c == 0) ? CC-1 : c-1;
            int o = f*STR+c, om = f*STR+cm;
            sZ[o] = mu1*sX[o] + sG1[o] + mu2*(sU[om]-sU[o]) + (sG2[om]-sG2[o]);
        }
        __syncthreads();

        // Phase C: Z = Minv (circular-conv) rhs, via fp32 WMMA 16x16x4, K padded to 32
        {
            const int fb = wv * 16;                    // this wave's fiber tile
            const int arow = (fb + laneM) * STR;
            v2f a[8];
#pragma unroll
            for (int kk = 0; kk < 8; ++kk) {           // A 16x4: v0=K0|K2, v1=K1|K3
                int k0 = 4*kk + 2*hi;
                v2f t; t.x = sZ[arow + k0]; t.y = sZ[arow + k0 + 1];
                a[kk] = t;
            }
            const float* mrow = &sM[it*32];
            v8f acc0 = {0.f,0.f,0.f,0.f,0.f,0.f,0.f,0.f};
            v8f acc1 = acc0;
#pragma unroll
            for (int kk = 0; kk < 8; ++kk) {
                int k0 = 4*kk + hi;                    // B 4x16: v0 = rows K0|K1
                int k1 = 4*kk + 2 + hi;                //          v1 = rows K2|K3
                v2f b0, b1;
                b0.x = bval(mrow, k0, laneM);
                b0.y = bval(mrow, k1, laneM);
                b1.x = bval(mrow, k0, 16 + laneM);
                b1.y = bval(mrow, k1, 16 + laneM);
                acc0 = __builtin_amdgcn_wmma_f32_16x16x4_f32(
                           false, a[kk], false, b0, (short)0, acc0, false, false);
                acc1 = __builtin_amdgcn_wmma_f32_16x16x4_f32(
                           false, a[kk], false, b1, (short)0, acc1, false, false);
            }
            // D 16x16: lane holds (M = r + 8*hi, N = laneM) in vgpr r
#pragma unroll
            for (int r = 0; r < 8; ++r) {
                int fr = fb + r + 8*hi;
                sZ[fr*STR + laneM]      = acc0[r];
                sZ[fr*STR + 16 + laneM] = acc1[r];   // col 31 is exact 0 -> keeps pad
            }
        }
        __syncthreads();

        // Phase D: DsZ = diff_h(Z); X update; G1 update; G2 update
        for (int i = tid; i < FPB*CC; i += TPB) {
            int f = i / CC, c = i - f*CC; int o = f*STR+c;
            int cp = (c == CC-1) ? 0 : c+1;
            float z   = sZ[o];
            float dsz = sZ[f*STR+cp] - z;
            float w = sWt[o], y = sY[o], g1 = sG1[o];
            float x = (w*y + mu1*z - g1) / (w + mu1);
            g1 = g1 + mu1*(x - z);
            sX[o] = x; sG1[o] = g1;
            sD[o] = dsz;
            sG2[o] = sG2[o] + mu2*(sU[o] - dsz);
        }
        __syncthreads();
        mud *= 1.05;
    }

    // output Z
    for (int i = tid; i < FPB*CC; i += TPB) {
        int f = i / CC, c = i - f*CC;
        gOut[base + i] = sZ[f*STR + c];
    }
}

extern "C" void kernel_launch(void* const* d_in, const int* in_sizes, int n_in,
                              void* d_out, int out_size, void* d_ws, size_t ws_size,
                              hipStream_t stream) {
    const float* Y    = (const float*)d_in[0];
    const float* inW  = (const float*)d_in[1];
    const float* demo = (const float*)d_in[2];
    float* out = (float*)d_out;
    (void)in_sizes; (void)n_in; (void)out_size; (void)d_ws; (void)ws_size;

    dim3 grid(FIB / FPB);   // 512 workgroups, each owns 128 fibers for all 20 iterations
    hipLaunchKernelGGL(hwtv_admm_kernel, grid, dim3(TPB), 0, stream, Y, inW, demo, out);
}